// TransR_15771119911422
// MI455X (gfx1250) — compile-verified
//
#include <hip/hip_runtime.h>
#include <hip/hip_bf16.h>

// ---------------------------------------------------------------------------
// TransR margin loss on MI455X (gfx1250), fp32 via V_WMMA_F32_16X16X4_F32.
//
// diff = (s_emb - t_emb) @ M + r_emb ; dist = sum|diff| ; loss = mean(relu(pos-neg+1))
//
// Pipeline (all on `stream`):
//   K0 zero     : zero relation counters/cursors and the output scalar
//   K1 hist     : histogram of r over 500 relations
//   K2 scan     : exclusive prefix sum -> group offsets
//   K3 scatter  : bucket sort of sample indices by relation
//   K4 main     : 1 workgroup per relation; M staged once into LDS in
//                 *B-fragment order* (one ds_load_b64 per WMMA B operand),
//                 batched U(16x128)@M(128x128) with wave32 WMMA 16x16x4 f32,
//                 global_prefetch of next block's node rows, per-row L1
//                 reduction, atomic accumulation of loss/BATCH into d_out[0].
// ---------------------------------------------------------------------------

#define DIMK        128
#define LINKS       500
#define BATCH       16384
#define MARGIN_F    1.0f

#define U_STRIDE    132   // LDS dword stride per U row (conflict-free A frags)
#define KP_STRIDE   288   // LDS dword stride per k-pair of swizzled M:
                          // 256 data dwords + 32 pad => kh=1 lanes land on
                          // banks 32..63 while kh=0 lanes use 0..31.

typedef float v2f __attribute__((ext_vector_type(2)));
typedef float v8f __attribute__((ext_vector_type(8)));

// ---------------- K0: zero counters + output ----------------
__global__ void transr_zero_kernel(int* counts, int* cursor, float* out) {
    int t = threadIdx.x;
    if (t < LINKS)                 counts[t] = 0;
    else if (t < 2 * LINKS)        cursor[t - LINKS] = 0;
    if (t == blockDim.x - 1)       out[0] = 0.0f;
}

// ---------------- K1: histogram of relations ----------------
__global__ void transr_hist_kernel(const int* __restrict__ r, int* __restrict__ counts) {
    int b = blockIdx.x * blockDim.x + threadIdx.x;
    if (b < BATCH) atomicAdd(&counts[r[b]], 1);
}

// ---------------- K2: exclusive scan over 500 bins (one block of 512) -------
__global__ void transr_scan_kernel(const int* __restrict__ counts, int* __restrict__ offsets) {
    __shared__ int tmp[512];
    int t = threadIdx.x;
    int c = (t < LINKS) ? counts[t] : 0;
    tmp[t] = c;
    __syncthreads();
    #pragma unroll
    for (int off = 1; off < 512; off <<= 1) {
        int v = (t >= off) ? tmp[t - off] : 0;
        __syncthreads();
        tmp[t] += v;
        __syncthreads();
    }
    if (t < LINKS) offsets[t] = tmp[t] - c;       // exclusive
    if (t == LINKS - 1) offsets[LINKS] = tmp[t];  // total == BATCH
}

// ---------------- K3: scatter sample ids into relation buckets --------------
__global__ void transr_scatter_kernel(const int* __restrict__ r,
                                      const int* __restrict__ offsets,
                                      int* __restrict__ cursor,
                                      int* __restrict__ sorted) {
    int b = blockIdx.x * blockDim.x + threadIdx.x;
    if (b < BATCH) {
        int rel = r[b];
        int p = offsets[rel] + atomicAdd(&cursor[rel], 1);
        sorted[p] = b;
    }
}

// ---------------- K4: main WMMA kernel, one block per relation --------------
__global__ void __launch_bounds__(256)
transr_main_kernel(const float* __restrict__ node_w,
                   const float* __restrict__ link_w,
                   const float* __restrict__ transfer_w,
                   const int* __restrict__ sp, const int* __restrict__ tp,
                   const int* __restrict__ sn, const int* __restrict__ tn,
                   const int* __restrict__ offsets,
                   const int* __restrict__ sorted,
                   float* __restrict__ out) {
    extern __shared__ float smem[];
    float* Mswz   = smem;                          // 64 k-pairs x KP_STRIDE
    float* Ulds   = Mswz + 64 * KP_STRIDE;         // 16 x U_STRIDE
    float* Remb   = Ulds + 16 * U_STRIDE;          // 128
    float* rowsum = Remb + DIMK;                   // 16

    const int rel   = blockIdx.x;
    const int tid   = threadIdx.x;
    const int start = offsets[rel];
    const int end   = offsets[rel + 1];
    if (start >= end) return;                      // uniform exit: empty bucket

    // ---- Stage M into LDS in B-fragment order --------------------------
    // Mswz[kp*KP_STRIDE + 2*col + j] = M[2*kp + j][col], kp = k/2.
    // Each WMMA B operand then reads one contiguous b64 per lane.
    {
        const float* Msrc = transfer_w + (size_t)rel * (DIMK * DIMK);
        for (int i = tid; i < (DIMK / 2) * DIMK; i += 256) {  // 8192 pairs
            int col = i & (DIMK - 1);
            int kp  = i >> 7;                                  // 0..63
            v2f p;
            p.x = Msrc[(size_t)(2 * kp)     * DIMK + col];
            p.y = Msrc[(size_t)(2 * kp + 1) * DIMK + col];
            *(v2f*)(Mswz + kp * KP_STRIDE + 2 * col) = p;
        }
        for (int i = tid; i < DIMK; i += 256)
            Remb[i] = link_w[(size_t)rel * DIMK + i];
    }
    __syncthreads();

    const int lane = tid & 31;
    const int wave = tid >> 5;                     // 8 waves -> 8 column tiles
    const int m    = lane & 15;                    // A-frag row
    const int kh   = lane >> 4;                    // K-half selector (0/1)
    const int col  = (wave << 4) + (lane & 15);    // output column (0..127)
    const float* Urow  = Ulds + m * U_STRIDE;
    const float* Bbase = Mswz + kh * KP_STRIDE + 2 * col;  // chunk c0 -> +c0*2*KP_STRIDE
    const float  rv    = Remb[col];

    // ---- Process this relation's samples 8 at a time (16 WMMA rows) -------
    for (int base = start; base < end; base += 8) {
        const int nb = min(8, end - base);

        // Prefetch next block's node rows (global_prefetch_b8) while this
        // block's WMMA chain runs. Each row is 512 B; touch it twice.
        {
            int nbase = base + 8;
            if (nbase < end && tid < 32) {
                int item = tid >> 2;               // 0..7
                if (nbase + item < end) {
                    int b   = sorted[nbase + item];
                    int q   = tid & 3;
                    int idx = (q == 0) ? sp[b] : (q == 1) ? tp[b]
                            : (q == 2) ? sn[b] : tn[b];
                    const float* row = node_w + (size_t)idx * DIMK;
                    __builtin_prefetch(row, 0, 3);
                    __builtin_prefetch(row + 64, 0, 3);
                }
            }
        }

        // Build U rows: row 2i = pos diff, row 2i+1 = neg diff; zero padding.
        for (int i = tid; i < 16 * DIMK; i += 256) {
            int row  = i >> 7;
            int c    = i & (DIMK - 1);
            int item = row >> 1;
            float val = 0.0f;
            if (item < nb) {
                int b = sorted[base + item];
                if (row & 1)
                    val = node_w[(size_t)sn[b] * DIMK + c] - node_w[(size_t)tn[b] * DIMK + c];
                else
                    val = node_w[(size_t)sp[b] * DIMK + c] - node_w[(size_t)tp[b] * DIMK + c];
            }
            Ulds[row * U_STRIDE + c] = val;
        }
        if (tid < 16) rowsum[tid] = 0.0f;
        __syncthreads();

        // U(16x128) @ M(128x128): wave handles one 16x16 output tile,
        // 32 chained V_WMMA_F32_16X16X4_F32 over K.
        // Per WMMA: one ds_load_2addr_b32 (A) + one ds_load_b64 (B).
        v8f acc = {};
        #pragma unroll
        for (int c0 = 0; c0 < DIMK / 4; ++c0) {
            const int k = (c0 << 2) + (kh << 1);
            v2f a;  a.x = Urow[k];  a.y = Urow[k + 1];
            v2f bf = *(const v2f*)(Bbase + c0 * (2 * KP_STRIDE));
            acc = __builtin_amdgcn_wmma_f32_16x16x4_f32(
                false, a, false, bf, (short)0, acc, false, false);
        }

        // Epilogue: diff = acc + r_emb[col]; |.|; reduce over this wave's
        // 16 columns via shuffle butterfly, then LDS-atomic into rowsum.
        #pragma unroll
        for (int v = 0; v < 8; ++v) {
            float d = fabsf(acc[v] + rv);
            #pragma unroll
            for (int off = 1; off < 16; off <<= 1)
                d += __shfl_xor(d, off, 32);
            if ((lane & 15) == 0)
                atomicAdd(&rowsum[v + (kh << 3)], d);   // row = v + 8*kh
        }
        __syncthreads();

        // Margin loss per sample, pre-scaled by 1/BATCH.
        if (tid < nb) {
            float pos = rowsum[2 * tid];
            float neg = rowsum[2 * tid + 1];
            float l   = pos - neg + MARGIN_F;
            if (l > 0.0f) atomicAdd(out, l * (1.0f / (float)BATCH));
        }
        __syncthreads();   // before next iteration overwrites Ulds/rowsum
    }
}

// ---------------------------------------------------------------------------
extern "C" void kernel_launch(void* const* d_in, const int* in_sizes, int n_in,
                              void* d_out, int out_size, void* d_ws, size_t ws_size,
                              hipStream_t stream) {
    const float* node_w     = (const float*)d_in[0];
    const float* link_w     = (const float*)d_in[1];
    const float* transfer_w = (const float*)d_in[2];
    const int*   sp         = (const int*)d_in[3];
    const int*   tp         = (const int*)d_in[4];
    const int*   sn         = (const int*)d_in[5];
    const int*   tn         = (const int*)d_in[6];
    const int*   r          = (const int*)d_in[7];
    float*       out        = (float*)d_out;

    // workspace layout (ints): counts[512] | cursor[512] | offsets[512] | sorted[BATCH]
    int* counts  = (int*)d_ws;
    int* cursor  = counts + 512;
    int* offsets = cursor + 512;
    int* sorted  = offsets + 512;

    transr_zero_kernel<<<1, 1024, 0, stream>>>(counts, cursor, out);
    transr_hist_kernel<<<BATCH / 256, 256, 0, stream>>>(r, counts);
    transr_scan_kernel<<<1, 512, 0, stream>>>(counts, offsets);
    transr_scatter_kernel<<<BATCH / 256, 256, 0, stream>>>(r, offsets, cursor, sorted);

    const size_t lds_bytes =
        (size_t)(64 * KP_STRIDE + 16 * U_STRIDE + DIMK + 16) * sizeof(float); // 82752 B
    transr_main_kernel<<<LINKS, 256, lds_bytes, stream>>>(
        node_w, link_w, transfer_w, sp, tp, sn, tn, offsets, sorted, out);
}